// KANC_CIFAR_34703335752282
// MI455X (gfx1250) — compile-verified
//
#include <hip/hip_runtime.h>

typedef __attribute__((ext_vector_type(2))) float v2f;
typedef __attribute__((ext_vector_type(8))) float v8f;

#define PIX_PER_WG 128
#define CHUNK_F    8
#define CHUNK_K    56          // CHUNK_F * 7
#define A_STRIDE   58          // even -> 8B-aligned v2f LDS loads; 58*m mod 64 distinct for m<16
#define WT_STRIDE  58          // transposed weight slab stride (same property)

// ---------------------------------------------------------------------------
// Uniform-grid cubic B-spline bases, grid_size=3, order=3, range [-1,1]:
// extended grid g[i] = -3 + i*(2/3), i=0..9.  Produces 6 bases (Cox-de-Boor).
// Matches reference b_splines() exactly (half-open [g_i, g_{i+1}) base case).
// ---------------------------------------------------------------------------
__device__ __forceinline__ void kan_row7(float v, float* out7) {
    // out7[0] = silu(v); out7[1..6] = B0..B5(v)
    out7[0] = v / (1.0f + __expf(-v));
    const float h = 2.0f / 3.0f;
    float g[10];
#pragma unroll
    for (int i = 0; i < 10; ++i) g[i] = -3.0f + h * (float)i;
    float b[9];
#pragma unroll
    for (int i = 0; i < 9; ++i) b[i] = (v >= g[i] && v < g[i + 1]) ? 1.0f : 0.0f;
#pragma unroll
    for (int j = 1; j <= 3; ++j) {
        const float inv = 1.0f / (h * (float)j);   // uniform grid: all denoms = j*h
#pragma unroll
        for (int i = 0; i < 9 - j; ++i) {
            float left  = (v - g[i]) * inv;
            float right = (g[i + j + 1] - v) * inv;
            b[i] = left * b[i] + right * b[i + 1];
        }
    }
#pragma unroll
    for (int i = 0; i < 6; ++i) out7[1 + i] = b[i];
}

// ---------------------------------------------------------------------------
// Fold (base_w, spline_w, scaler) into padded GEMM weight matrix W[Kpad][OP]:
//   k = f*7 + j :  j==0 -> base_w[o,f] ; j=1..6 -> spline_w[o,f,j-1]*scaler[o,f]
// Zero-filled for f>=F (feature padding) and n>=O (channel padding).
// ---------------------------------------------------------------------------
__global__ void kan_prep_weights(const float* __restrict__ base_w,
                                 const float* __restrict__ spline_w,
                                 const float* __restrict__ scaler,
                                 float* __restrict__ Wmat,
                                 int F, int Fpad, int O, int OP) {
    int idx = blockIdx.x * blockDim.x + threadIdx.x;
    int total = Fpad * 7 * OP;
    if (idx >= total) return;
    int k = idx / OP, n = idx % OP;
    int f = k / 7,  j = k % 7;
    float w = 0.0f;
    if (f < F && n < O) {
        if (j == 0) w = base_w[n * F + f];
        else        w = spline_w[(n * F + f) * 6 + (j - 1)] * scaler[n * F + f];
    }
    Wmat[idx] = w;
}

// ---------------------------------------------------------------------------
// Fused KAN 3x3 conv layer as WMMA GEMM (f32 16x16x4).
// Workgroup: 256 threads = 8 waves, 128 pixels. Wave w owns pixel rows
// [16w,16w+16) of the block. NT = OP/16 column tiles (1 or 2).
// A fragment (lane l): M = l%16; v2f = A[K + {0,1}] (l<16) or A[K + {2,3}].
// B fragment mirrored with N = l%16 -> both are single ds_load_b64 from LDS.
// ---------------------------------------------------------------------------
template <int NT>
__global__ __launch_bounds__(256)
void kan_conv_wmma(const float* __restrict__ in, const float* __restrict__ Wmat,
                   float* __restrict__ out,
                   int C, int H, int W, int O, int Fpad) {
    constexpr int OP = NT * 16;
    __shared__ __align__(16) float Alds[PIX_PER_WG * A_STRIDE];
    __shared__ __align__(16) float WldsT[OP * WT_STRIDE];   // transposed: [n][k]

    const int tid  = threadIdx.x;
    const int lane = tid & 31;
    const int wave = tid >> 5;
    const int l16  = lane & 15;
    const int hi2  = (lane >= 16) ? 2 : 0;
    const int pixBase = blockIdx.x * PIX_PER_WG;
    const int HW = H * W;
    const int F  = C * 9;
    const int nChunks = Fpad / CHUNK_F;

    v8f acc0 = {};
    v8f acc1 = {};

    for (int cf = 0; cf < nChunks; ++cf) {
        // ---- stage weight slab W[cf*56 .. +56)[0..OP) into LDS, transposed ----
        for (int i = tid; i < CHUNK_K * OP; i += 256) {
            int k = i / OP, n = i - k * OP;
            WldsT[n * WT_STRIDE + k] = Wmat[cf * CHUNK_K * OP + i];
        }

        // ---- compute A slab: 128 pixels x 8 features -> 56 K-columns ----
        for (int p = tid; p < PIX_PER_WG * CHUNK_F; p += 256) {
            int pix = p >> 3;
            int fi  = p & 7;
            int f   = cf * CHUNK_F + fi;
            int n   = pixBase + pix;
            int b   = n / HW;
            int rem = n - b * HW;
            int y   = rem / W;
            int x   = rem - y * W;
            float v = 0.0f;
            if (f < F) {
                int c  = f / 9;
                int t  = f - c * 9;
                int iy = y + t / 3 - 1;
                int ix = x + t % 3 - 1;
                if (iy >= 0 && iy < H && ix >= 0 && ix < W)
                    v = in[((b * C + c) * H + iy) * W + ix];
            }
            kan_row7(v, &Alds[pix * A_STRIDE + fi * 7]);
        }
        __syncthreads();

        // ---- 14 WMMA K-steps over this slab ----
        const int arow = (wave * 16 + l16) * A_STRIDE + hi2;   // lane's A base
        const int b0row = l16 * WT_STRIDE + hi2;               // col tile 0
        const int b1row = (16 + l16) * WT_STRIDE + hi2;        // col tile 1
#pragma unroll
        for (int kb = 0; kb < CHUNK_K; kb += 4) {
            v2f a  = *(const v2f*)&Alds[arow + kb];
            v2f b0 = *(const v2f*)&WldsT[b0row + kb];
            acc0 = __builtin_amdgcn_wmma_f32_16x16x4_f32(
                false, a, false, b0, (short)0, acc0, false, false);
            if (NT > 1) {
                v2f b1 = *(const v2f*)&WldsT[b1row + kb];
                acc1 = __builtin_amdgcn_wmma_f32_16x16x4_f32(
                    false, a, false, b1, (short)0, acc1, false, false);
            }
        }
        __syncthreads();
    }

    // ---- store D tiles to out[B,O,H,W] ----
#pragma unroll
    for (int r = 0; r < 8; ++r) {
        int m   = pixBase + wave * 16 + r + (hi2 ? 8 : 0);   // global pixel index
        int b   = m / HW;
        int rem = m - b * HW;
        int y   = rem / W;
        int x   = rem - y * W;
        if (l16 < O)
            out[((b * O + l16) * H + y) * W + x] = acc0[r];
        if (NT > 1 && (16 + l16) < O)
            out[((b * O + 16 + l16) * H + y) * W + x] = acc1[r];
    }
}

// ---------------------------------------------------------------------------
// 2x2 max-pool, stride 2 (windows always in-bounds: H,W even).
// ---------------------------------------------------------------------------
__global__ void kan_maxpool2(const float* __restrict__ in, float* __restrict__ out,
                             int BC, int H, int W) {
    int Ho = H >> 1, Wo = W >> 1;
    int idx = blockIdx.x * blockDim.x + threadIdx.x;
    if (idx >= BC * Ho * Wo) return;
    int wo = idx % Wo;
    int t  = idx / Wo;
    int ho = t % Ho;
    int bc = t / Ho;
    const float* p = in + (bc * H + ho * 2) * W + wo * 2;
    out[idx] = fmaxf(fmaxf(p[0], p[1]), fmaxf(p[W], p[W + 1]));
}

// ---------------------------------------------------------------------------
// Final FC: out[256,100] = h[256,512] @ lin_w[100,512]^T + lin_b.
// 16 row tiles x 7 col tiles (100 padded to 112), K=512 in WMMA steps of 4.
// 14 blocks x 8 waves = exactly 112 wave-tiles. Column padding handled with a
// clamped index + 0/1 mask multiply (no EXEC divergence in the K loop).
// ---------------------------------------------------------------------------
__global__ __launch_bounds__(256)
void kan_fc_wmma(const float* __restrict__ h, const float* __restrict__ lin_w,
                 const float* __restrict__ lin_b, float* __restrict__ out) {
    const int tid  = threadIdx.x;
    const int lane = tid & 31;
    const int wave = tid >> 5;
    const int gw   = blockIdx.x * 8 + wave;       // 0..111
    const int rt   = gw / 7;
    const int ct   = gw % 7;
    const int l16  = lane & 15;
    const int hi2  = (lane >= 16) ? 2 : 0;
    const int m    = rt * 16 + l16;               // row for A fragment
    const int n    = ct * 16 + l16;               // col for B fragment
    const int nc   = (n < 100) ? n : 99;          // clamped (valid) index
    const float msk = (n < 100) ? 1.0f : 0.0f;

    const float* hrow = h + m * 512 + hi2;
    const float* wrow = lin_w + nc * 512 + hi2;

    v8f acc = {};
#pragma unroll 4
    for (int kb = 0; kb < 512; kb += 4) {
        v2f a  = *(const v2f*)&hrow[kb];
        v2f bm = *(const v2f*)&wrow[kb];
        bm.x *= msk;
        bm.y *= msk;
        acc = __builtin_amdgcn_wmma_f32_16x16x4_f32(
            false, a, false, bm, (short)0, acc, false, false);
    }
    float bias = lin_b[nc];
#pragma unroll
    for (int r = 0; r < 8; ++r) {
        int row = rt * 16 + r + (hi2 ? 8 : 0);
        if (n < 100) out[row * 100 + n] = acc[r] + bias;
    }
}

// ---------------------------------------------------------------------------
extern "C" void kernel_launch(void* const* d_in, const int* in_sizes, int n_in,
                              void* d_out, int out_size, void* d_ws, size_t ws_size,
                              hipStream_t stream) {
    const float* x         = (const float*)d_in[0];   // [256,3,32,32]
    const float* c1_base_w = (const float*)d_in[1];   // [8,27]
    const float* c1_spl_w  = (const float*)d_in[2];   // [8,27,6]
    const float* c1_scaler = (const float*)d_in[3];   // [8,27]
    const float* c2_base_w = (const float*)d_in[4];   // [16,72]
    const float* c2_spl_w  = (const float*)d_in[5];   // [16,72,6]
    const float* c2_scaler = (const float*)d_in[6];   // [16,72]
    const float* c3_base_w = (const float*)d_in[7];   // [32,144]
    const float* c3_spl_w  = (const float*)d_in[8];   // [32,144,6]
    const float* c3_scaler = (const float*)d_in[9];   // [32,144]
    const float* lin_w     = (const float*)d_in[10];  // [100,512]
    const float* lin_b     = (const float*)d_in[11];  // [100]
    float* out             = (float*)d_out;           // [256,100]

    // Workspace layout (floats); total ~12.2 MB
    float* ws = (float*)d_ws;
    float* W1    = ws;                 // 224*16   = 3584
    float* W2    = W1 + 224 * 16;      // 504*16   = 8064
    float* W3    = W2 + 504 * 16;      // 1008*32  = 32256
    float* convB = W3 + 1008 * 32;     // max conv out: 256*8*32*32 = 2097152
    float* pool1 = convB + 2097152;    // 256*8*16*16  = 524288
    float* pool2 = pool1 + 524288;     // 256*16*8*8   = 262144
    float* pool3 = pool2 + 262144;     // 256*32*4*4   = 131072
    (void)ws_size; (void)n_in; (void)in_sizes; (void)out_size;

    // ---- fold weights into GEMM form ----
    kan_prep_weights<<<(224 * 16 + 255) / 256, 256, 0, stream>>>(
        c1_base_w, c1_spl_w, c1_scaler, W1, 27, 32, 8, 16);
    kan_prep_weights<<<(504 * 16 + 255) / 256, 256, 0, stream>>>(
        c2_base_w, c2_spl_w, c2_scaler, W2, 72, 72, 16, 16);
    kan_prep_weights<<<(1008 * 32 + 255) / 256, 256, 0, stream>>>(
        c3_base_w, c3_spl_w, c3_scaler, W3, 144, 144, 32, 32);

    // ---- layer 1: [256,3,32,32] -> conv [256,8,32,32] -> pool [256,8,16,16]
    kan_conv_wmma<1><<<(256 * 32 * 32) / PIX_PER_WG, 256, 0, stream>>>(
        x, W1, convB, 3, 32, 32, 8, 32);
    kan_maxpool2<<<(256 * 8 * 16 * 16 + 255) / 256, 256, 0, stream>>>(
        convB, pool1, 256 * 8, 32, 32);

    // ---- layer 2: -> conv [256,16,16,16] -> pool [256,16,8,8]
    kan_conv_wmma<1><<<(256 * 16 * 16) / PIX_PER_WG, 256, 0, stream>>>(
        pool1, W2, convB, 8, 16, 16, 16, 72);
    kan_maxpool2<<<(256 * 16 * 8 * 8 + 255) / 256, 256, 0, stream>>>(
        convB, pool2, 256 * 16, 16, 16);

    // ---- layer 3: -> conv [256,32,8,8] -> pool [256,32,4,4] (== h[256,512])
    kan_conv_wmma<2><<<(256 * 8 * 8) / PIX_PER_WG, 256, 0, stream>>>(
        pool2, W3, convB, 16, 8, 8, 32, 144);
    kan_maxpool2<<<(256 * 32 * 4 * 4 + 255) / 256, 256, 0, stream>>>(
        convB, pool3, 256 * 32, 8, 8);

    // ---- FC: [256,512] @ [100,512]^T + b -> [256,100]
    kan_fc_wmma<<<14, 256, 0, stream>>>(pool3, lin_w, lin_b, out);
}